// LIFNode_57501022159089
// MI455X (gfx1250) — compile-verified
//
#include <hip/hip_runtime.h>
#include <cstdint>
#include <cstddef>

// LIF recurrence, B=32, T=64, N=32768, fp32.
// Memory-bound streaming kernel: 512MB @ 23.3TB/s ~= 22us floor.
// CDNA5 path: 4-deep global_load_async_to_lds_b128 pipeline (ASYNCcnt),
// ds_load_b128 reads, non-temporal b128 stores.

typedef float v4f __attribute__((ext_vector_type(4)));

#define TPB     256
#define DEPTH   4      // async tiles in flight
#define SLOTS   8      // LDS ring slots (2x depth: kills LDS WAR hazard)
#define T_STEPS 64
#define N_ELEM  32768
#define N4      (N_ELEM / 4)     // 8192 float4 per (b,t) row
#define B_DIM   32

#define INV_TAU 0.2f    // 1/5.0
#define ALPHA_C 0.1f
#define BETA_C  0.5f
#define VTH_C   0.5f
// V_RESET == 0.0 -> hard reset is mem *= (1 - spike)

__device__ __forceinline__ void lif_step4(const v4f x, v4f& mem, v4f& cache, v4f& sp) {
#pragma unroll
    for (int i = 0; i < 4; ++i) {
        float xv = x[i];
        float m  = mem[i];
        float c  = cache[i];
        float drive = __builtin_fmaf(ALPHA_C * c, xv, xv);   // (1 + a*c) * x
        m = __builtin_fmaf(m, INV_TAU, drive);               // mem/tau + drive
        float s  = (m > VTH_C) ? 1.0f : 0.0f;                // heaviside spike
        float ns = 1.0f - s;
        m = m * ns;                                          // hard reset to 0
        c = __builtin_fmaf(BETA_C, c, ns);                   // beta*c + (1-s)
        mem[i]   = m;
        cache[i] = c;
        sp[i]    = s;
    }
}

__device__ __forceinline__ void async_load_tile(const v4f* gp, uint32_t lds_off) {
    // LDS destination address comes from lds_off; tracked by ASYNCcnt.
    asm volatile("global_load_async_to_lds_b128 %0, %1, off th:TH_LOAD_NT"
                 :: "v"(lds_off), "v"(gp)
                 : "memory");
}

__global__ __launch_bounds__(TPB) void lif_scan_kernel(const v4f* __restrict__ X,
                                                       v4f* __restrict__ out) {
    __shared__ v4f buf[SLOTS][TPB];   // 32 KB ring

    const int tid = threadIdx.x;
    const size_t g  = (size_t)blockIdx.x * TPB + tid;   // [0, B*N4)
    const size_t b  = g >> 13;          // g / N4
    const size_t n4 = g & (N4 - 1);     // g % N4
    const size_t base = b * ((size_t)T_STEPS * N4) + n4;   // float4 units

    // Prime the async pipeline: tiles t = 0 .. DEPTH-1 in flight.
#pragma unroll
    for (int t = 0; t < DEPTH; ++t) {
        uint32_t lds_off = (uint32_t)(uintptr_t)(&buf[t & (SLOTS - 1)][tid]);
        async_load_tile(X + base + (size_t)t * N4, lds_off);
    }

    v4f mem   = {0.0f, 0.0f, 0.0f, 0.0f};
    v4f cache = {0.0f, 0.0f, 0.0f, 0.0f};

    // Main loop: consume tile t, refill tile t+DEPTH.
    for (int t = 0; t < T_STEPS - DEPTH; ++t) {
        // Async loads complete in order: <=DEPTH-1 outstanding => tile t landed.
        asm volatile("s_wait_asynccnt 0x3" ::: "memory");

        v4f x = buf[t & (SLOTS - 1)][tid];     // ds_load_b128
        v4f sp;
        lif_step4(x, mem, cache, sp);
        __builtin_nontemporal_store(sp, out + base + (size_t)t * N4);

        const int tn = t + DEPTH;
        uint32_t lds_off = (uint32_t)(uintptr_t)(&buf[tn & (SLOTS - 1)][tid]);
        async_load_tile(X + base + (size_t)tn * N4, lds_off);
    }

    // Tail: everything issued; drain and finish the last DEPTH steps.
    asm volatile("s_wait_asynccnt 0x0" ::: "memory");
#pragma unroll
    for (int t = T_STEPS - DEPTH; t < T_STEPS; ++t) {
        v4f x = buf[t & (SLOTS - 1)][tid];
        v4f sp;
        lif_step4(x, mem, cache, sp);
        __builtin_nontemporal_store(sp, out + base + (size_t)t * N4);
    }
}

extern "C" void kernel_launch(void* const* d_in, const int* in_sizes, int n_in,
                              void* d_out, int out_size, void* d_ws, size_t ws_size,
                              hipStream_t stream) {
    (void)in_sizes; (void)n_in; (void)d_ws; (void)ws_size; (void)out_size;
    const v4f* X  = (const v4f*)d_in[0];   // [B, T, N] fp32
    v4f* out      = (v4f*)d_out;           // [B, T, N] fp32 spikes

    const int total_threads = B_DIM * N4;          // 262144
    dim3 grid(total_threads / TPB);                // 1024 blocks
    lif_scan_kernel<<<grid, TPB, 0, stream>>>(X, out);
}